// PolarizationSolver_4002909520035
// MI455X (gfx1250) — compile-verified
//
#include <hip/hip_runtime.h>
#include <math.h>

#define A_DAMP_F 0.39f
#define CHG_FACTOR_F 1389.35457644382f

constexpr int B = 4, N = 256, C = 8;
constexpr int NS = B * C;           // 32 systems
constexpr int NDIM = 3 * N;         // 768
constexpr int NBLK = NDIM / 16;     // 48 tile-rows
constexpr long long NN = (long long)N * N;

// ---- output offsets (floats), concatenated in reference return order ----
constexpr long long OFF_EP = 0;
constexpr long long OFF_MU = OFF_EP + (long long)B * C * NDIM;       // 24576
constexpr long long OFF_E  = OFF_MU + (long long)B * N * C * 3;      // 49152
constexpr long long OFF_F  = OFF_E  + (long long)B * C;              // 49184
constexpr long long OFF_S  = OFF_F  + (long long)B * N * C * 3;      // 73760
constexpr long long OFF_D1 = OFF_S  + (long long)B * C * NN;
constexpr long long OFF_D2 = OFF_D1 + (long long)B * C * NN;

// ---- workspace offsets (floats) ----
constexpr long long WS_SC  = 0;                                   // B*N*N
constexpr long long WS_A   = WS_SC  + (long long)B * NN;          // NS*768*768
constexpr long long WS_IL  = WS_A   + (long long)NS * NDIM * NDIM;// NS*NBLK*256
constexpr long long WS_MU  = WS_IL  + (long long)NS * NBLK * 256; // NS*768

typedef float v2f __attribute__((ext_vector_type(2)));
typedef float v8f __attribute__((ext_vector_type(8)));

// ================= scaler construction =================
__global__ void pol_scaler_init(float* __restrict__ sc) {
  long long idx = (long long)blockIdx.x * blockDim.x + threadIdx.x;
  if (idx >= (long long)B * NN) return;
  int j = (int)(idx % N);
  int i = (int)((idx / N) % N);
  sc[idx] = (i == j) ? 0.0f : 1.0f;
}

__global__ void pol_scaler_pairs(float* __restrict__ sc, const int* __restrict__ idx,
                                 int npairs, const float* __restrict__ valp) {
  int t = blockIdx.x * blockDim.x + threadIdx.x;
  if (t >= npairs) return;
  int a0 = idx[2 * t], a1 = idx[2 * t + 1];
  int b = a0 / N, i = a0 % N, j = a1 % N;
  float v = valp ? valp[0] : 0.0f;
  sc[((long long)b * N + i) * N + j] = v;
  sc[((long long)b * N + j) * N + i] = v;
}

// ================= pair tensors: D1/D2/scaler outs + A matrix =================
__global__ __launch_bounds__(256) void pol_build(const float* __restrict__ coords,
                                                 const float* __restrict__ alpha,
                                                 const float* __restrict__ sc,
                                                 float* __restrict__ out,
                                                 float* __restrict__ Aw) {
  long long gid = (long long)blockIdx.x * 256 + threadIdx.x; // (b,c,i,j)
  if (gid >= (long long)B * C * NN) return;
  int j = (int)(gid % N);
  long long t = gid / N;
  int i = (int)(t % N); t /= N;
  int c = (int)(t % C);
  int b = (int)(t / C);

  const float* pi = coords + (((long long)(b * N + i)) * C + c) * 3;
  const float* pj = coords + (((long long)(b * N + j)) * C + c) * 3;
  float ux = pi[0] - pj[0], uy = pi[1] - pj[1], uz = pi[2] - pj[2];
  float nrm = sqrtf(ux * ux + uy * uy + uz * uz);
  float rn = fmaxf(nrm, 1e-3f);
  float ai = alpha[b * N + i], aj = alpha[b * N + j];
  float sa = sqrtf(ai * aj);
  float rn2 = rn * rn, rn3 = rn2 * rn, rn5 = rn3 * rn2;
  float d = fmaxf(-A_DAMP_F * rn3 / sa, -1e6f);
  float ed = expf(d);
  float D1 = 1.0f - ed;
  float D2 = fmaxf(1.0f - (1.0f - d) * ed, 1e-6f);
  float s = sc[((long long)b * N + i) * N + j];

  out[OFF_S  + gid] = s;
  out[OFF_D1 + gid] = D1;
  out[OFF_D2 + gid] = D2;

  int sys = b * C + c;
  float* Ab = Aw + (long long)sys * NDIM * NDIM + (long long)(3 * i) * NDIM + 3 * j;
  float u[3] = {ux, uy, uz};
  float inv5 = 1.0f / rn5;
  float invai = 1.0f / ai;
  #pragma unroll
  for (int x = 0; x < 3; ++x)
    #pragma unroll
    for (int y = 0; y < 3; ++y) {
      float Tv = -(3.0f * u[x] * u[y] * D2 - ((x == y) ? rn2 * D1 : 0.0f)) * inv5;
      if (i == j && x == y) Tv += invai;
      Ab[x * NDIM + y] = Tv;
    }
}

// ================= E_p field =================
__global__ __launch_bounds__(256) void pol_ep(const float* __restrict__ coords,
                                              const float* __restrict__ alpha,
                                              const float* __restrict__ charge,
                                              const float* __restrict__ sc,
                                              float* __restrict__ out) {
  int gid = blockIdx.x * 256 + threadIdx.x; // (b,c,i)
  if (gid >= B * C * N) return;
  int i = gid % N; int t = gid / N; int c = t % C; int b = t / C;
  const float* pi = coords + (((long long)(b * N + i)) * C + c) * 3;
  float ai = alpha[b * N + i];
  float ex = 0.0f, ey = 0.0f, ez = 0.0f;
  for (int j = 0; j < N; ++j) {
    const float* pj = coords + (((long long)(b * N + j)) * C + c) * 3;
    float ux = pi[0] - pj[0], uy = pi[1] - pj[1], uz = pi[2] - pj[2];
    float nrm = sqrtf(ux * ux + uy * uy + uz * uz);
    float rn = fmaxf(nrm, 1e-3f);
    float sa = sqrtf(ai * alpha[b * N + j]);
    float rn3 = rn * rn * rn;
    float d = fmaxf(-A_DAMP_F * rn3 / sa, -1e6f);
    float D1 = 1.0f - expf(d);
    float w = sc[((long long)b * N + i) * N + j] * charge[b * N + j] * D1 / rn3;
    ex += w * ux; ey += w * uy; ez += w * uz;
  }
  long long base = OFF_EP + ((long long)(b * C + c)) * NDIM + 3 * i;
  out[base + 0] = ex; out[base + 1] = ey; out[base + 2] = ez;
}

// ================= blocked WMMA Cholesky (fp32 16x16x4) =================
__global__ __launch_bounds__(256) void pol_chol(float* __restrict__ Aw,
                                                float* __restrict__ iLw) {
  int sys = blockIdx.x;
  float* Ag = Aw + (long long)sys * NDIM * NDIM;
  float* iLg = iLw + (long long)sys * NBLK * 256;
  __shared__ float Ld[256];
  __shared__ float Li[256];
  __shared__ float P[(NBLK - 1) * 256];
  const int tid  = threadIdx.x;
  const int w    = tid >> 5;
  const int lane = tid & 31;
  const int mrow = lane & 15;
  const int h    = lane >> 4;

  for (int k = 0; k < NBLK; ++k) {
    Ld[tid] = Ag[(k * 16 + (tid >> 4)) * NDIM + k * 16 + (tid & 15)];
    __syncthreads();
    if (tid == 0) {  // tiny sequential Cholesky of 16x16 diag block
      for (int cc = 0; cc < 16; ++cc) {
        float v = Ld[cc * 16 + cc];
        for (int q = 0; q < cc; ++q) v -= Ld[cc * 16 + q] * Ld[cc * 16 + q];
        float l = sqrtf(fmaxf(v, 1e-20f));
        Ld[cc * 16 + cc] = l;
        float inv = 1.0f / l;
        for (int r = cc + 1; r < 16; ++r) {
          float v2 = Ld[r * 16 + cc];
          for (int q = 0; q < cc; ++q) v2 -= Ld[r * 16 + q] * Ld[cc * 16 + q];
          Ld[r * 16 + cc] = v2 * inv;
        }
        for (int r = 0; r < cc; ++r) Ld[r * 16 + cc] = 0.0f; // clean upper
      }
    }
    __syncthreads();
    if (tid < 16) {  // column-parallel lower-triangular inverse
      int cc = tid;
      for (int r = 0; r < 16; ++r) Li[r * 16 + cc] = 0.0f;
      Li[cc * 16 + cc] = 1.0f / Ld[cc * 16 + cc];
      for (int r = cc + 1; r < 16; ++r) {
        float s = 0.0f;
        for (int q = cc; q < r; ++q) s += Ld[r * 16 + q] * Li[q * 16 + cc];
        Li[r * 16 + cc] = -s / Ld[r * 16 + r];
      }
    }
    __syncthreads();
    Ag[(k * 16 + (tid >> 4)) * NDIM + k * 16 + (tid & 15)] = Ld[tid];
    iLg[k * 256 + tid] = Li[tid];

    // panel: L_ik = M_ik * Li^T  (one wave per tile-row, WMMA f32 16x16x4 x4)
    for (int i = k + 1 + w; i < NBLK; i += 8) {
      v8f acc = {};
      const float* Arow = Ag + (i * 16 + mrow) * NDIM + k * 16;
      #pragma unroll
      for (int q = 0; q < 4; ++q) {
        int kk = 4 * q + 2 * h;
        v2f a  = *(const v2f*)(Arow + kk);            // A[m][kk..kk+1], m = mrow
        v2f bb = *(const v2f*)(Li + mrow * 16 + kk);  // B[kk][n] = Li[n][kk], n = mrow
        acc = __builtin_amdgcn_wmma_f32_16x16x4_f32(false, a, false, bb,
                                                    (short)0, acc, false, false);
      }
      #pragma unroll
      for (int v = 0; v < 8; ++v) {
        int mm = v + 8 * h;
        float val = acc[v];
        P[(i - k - 1) * 256 + mm * 16 + mrow] = val;
        Ag[(i * 16 + mm) * NDIM + k * 16 + mrow] = val;
      }
    }
    __threadfence_block();
    __syncthreads();

    // trailing update: M_ij -= L_ik * L_jk^T  (A negated on load; f32 WMMA NEG is C-only)
    int m = NBLK - 1 - k;
    int cnt = m * (m + 1) / 2;
    // incremental decode of flat triangular index t -> (ii, jj), jj <= ii
    int ii = 0, jj = w;
    while (jj > ii) { jj -= ii + 1; ++ii; }
    for (int t = w; t < cnt; t += 8) {
      int i = k + 1 + ii, j = k + 1 + jj;
      float* Cp = Ag + (long long)(i * 16) * NDIM + j * 16;
      v8f acc;
      #pragma unroll
      for (int v = 0; v < 8; ++v) acc[v] = Cp[(v + 8 * h) * NDIM + mrow];
      const float* Pa = P + ii * 256;
      const float* Pb = P + jj * 256;
      #pragma unroll
      for (int q = 0; q < 4; ++q) {
        int kk = 4 * q + 2 * h;
        v2f pa = *(const v2f*)(Pa + mrow * 16 + kk);
        v2f a;
        a[0] = -pa[0];
        a[1] = -pa[1];
        v2f bb = *(const v2f*)(Pb + mrow * 16 + kk);
        acc = __builtin_amdgcn_wmma_f32_16x16x4_f32(false, a, false, bb,
                                                    (short)0, acc, false, false);
      }
      #pragma unroll
      for (int v = 0; v < 8; ++v) Cp[(v + 8 * h) * NDIM + mrow] = acc[v];
      jj += 8;
      while (jj > ii) { jj -= ii + 1; ++ii; }
    }
    __threadfence_block();
    __syncthreads();
  }
}

// ================= blocked triangular solves + e_ind + mu_p =================
__global__ __launch_bounds__(256) void pol_solve(const float* __restrict__ Aw,
                                                 const float* __restrict__ iLw,
                                                 float* __restrict__ out,
                                                 float* __restrict__ muw) {
  int sys = blockIdx.x;
  const float* Ag  = Aw + (long long)sys * NDIM * NDIM;
  const float* iLg = iLw + (long long)sys * NBLK * 256;
  const float* Eg  = out + OFF_EP + (long long)sys * NDIM;
  __shared__ float yv[NDIM];
  __shared__ float z[16];
  __shared__ float red[256];
  int tid = threadIdx.x;
  int r = tid >> 4, cc = tid & 15;
  for (int idx = tid; idx < NDIM; idx += 256) yv[idx] = Eg[idx];
  __syncthreads();
  // forward: L y = E
  for (int ib = 0; ib < NBLK; ++ib) {
    float acc = 0.0f;
    const float* Lrow = Ag + (ib * 16 + r) * NDIM;
    for (int j = cc; j < ib * 16; j += 16) acc += Lrow[j] * yv[j];
    red[tid] = acc;
    __syncthreads();
    if (tid < 16) {
      float s = 0.0f;
      for (int q = 0; q < 16; ++q) s += red[tid * 16 + q];
      z[tid] = yv[ib * 16 + tid] - s;
    }
    __syncthreads();
    red[tid] = iLg[ib * 256 + r * 16 + cc] * z[cc];
    __syncthreads();
    if (tid < 16) {
      float s = 0.0f;
      for (int q = 0; q < 16; ++q) s += red[tid * 16 + q];
      yv[ib * 16 + tid] = s;
    }
    __syncthreads();
  }
  // backward: L^T mu = y  (in place: blocks > ib already hold mu)
  for (int ib = NBLK - 1; ib >= 0; --ib) {
    float acc = 0.0f;
    for (int j = (ib + 1) * 16 + cc; j < NDIM; j += 16)
      acc += Ag[(long long)j * NDIM + ib * 16 + r] * yv[j];
    red[tid] = acc;
    __syncthreads();
    if (tid < 16) {
      float s = 0.0f;
      for (int q = 0; q < 16; ++q) s += red[tid * 16 + q];
      z[tid] = yv[ib * 16 + tid] - s;
    }
    __syncthreads();
    red[tid] = iLg[ib * 256 + cc * 16 + r] * z[cc]; // invL^T
    __syncthreads();
    if (tid < 16) {
      float s = 0.0f;
      for (int q = 0; q < 16; ++q) s += red[tid * 16 + q];
      yv[ib * 16 + tid] = s;
    }
    __syncthreads();
  }
  int b = sys / C, c = sys % C;
  for (int idx = tid; idx < NDIM; idx += 256) {
    float m = yv[idx];
    muw[(long long)sys * NDIM + idx] = m;
    int i = idx / 3, x = idx % 3;
    out[OFF_MU + (((long long)(b * N + i)) * C + c) * 3 + x] = m;
  }
  float acc = 0.0f;
  for (int idx = tid; idx < NDIM; idx += 256) acc += yv[idx] * Eg[idx];
  red[tid] = acc;
  __syncthreads();
  for (int s2 = 128; s2 > 0; s2 >>= 1) {
    if (tid < s2) red[tid] += red[tid + s2];
    __syncthreads();
  }
  if (tid == 0) out[OFF_E + sys] = -0.5f * CHG_FACTOR_F * red[0];
}

// ================= analytic backward: f_ind =================
__global__ __launch_bounds__(256) void pol_grad(const float* __restrict__ coords,
                                                const float* __restrict__ alpha,
                                                const float* __restrict__ charge,
                                                const float* __restrict__ sc,
                                                const float* __restrict__ muw,
                                                float* __restrict__ out) {
  int gid = blockIdx.x * 256 + threadIdx.x; // (b,c,i)
  if (gid >= B * C * N) return;
  int i = gid % N; int t = gid / N; int c = t % C; int b = t / C;
  int sys = b * C + c;
  const float* pi = coords + (((long long)(b * N + i)) * C + c) * 3;
  const float* mi = muw + (long long)sys * NDIM + 3 * i;
  float mix = mi[0], miy = mi[1], miz = mi[2];
  float qi = charge[b * N + i], ai = alpha[b * N + i];
  float gx = 0.0f, gy = 0.0f, gz = 0.0f;
  for (int j = 0; j < N; ++j) {
    if (j == i) continue;
    const float* pj = coords + (((long long)(b * N + j)) * C + c) * 3;
    __builtin_prefetch(pj + 3 * C, 0, 1);
    float ux = pi[0] - pj[0], uy = pi[1] - pj[1], uz = pi[2] - pj[2];
    float nrm = sqrtf(ux * ux + uy * uy + uz * uz);
    if (nrm < 1e-3f) continue;  // clip(rn) active -> zero grad through pair
    float rn = nrm;
    float aj = alpha[b * N + j], qj = charge[b * N + j];
    float s = sc[((long long)b * N + i) * N + j];
    const float* mj = muw + (long long)sys * NDIM + 3 * j;
    float mjx = mj[0], mjy = mj[1], mjz = mj[2];
    float sa = sqrtf(ai * aj);
    float rn2 = rn * rn;
    float ir = 1.0f / rn, ir2 = ir * ir;
    float ir3 = ir * ir2, ir5 = ir3 * ir2, ir7 = ir5 * ir2;
    float draw = -A_DAMP_F * rn2 * rn / sa;
    float md = (draw >= -1e6f) ? 1.0f : 0.0f;
    float d = fmaxf(draw, -1e6f);
    float ed = expf(d);
    float D1 = 1.0f - ed;
    float D2r = 1.0f - (1.0f - d) * ed;
    float m2 = (D2r >= 1e-6f) ? 1.0f : 0.0f;
    float D2 = fmaxf(D2r, 1e-6f);
    float a  = mix * ux + miy * uy + miz * uz;   // mu_i . u
    float b2 = mjx * ux + mjy * uy + mjz * uz;   // mu_j . u
    float mm = mix * mjx + miy * mjy + miz * mjz;
    float g1 = s * qj, g2 = s * qi;
    float dde = md * d * ed;
    float cu = a * b2 * ir7 * (15.0f * D2 - 9.0f * m2 * md * d * d * ed)
             - 3.0f * ir5 * (dde + D1) * (mm - g1 * a + g2 * b2);
    float ci = -3.0f * D2 * b2 * ir5 - g1 * D1 * ir3;
    float cj = -3.0f * D2 * a * ir5 + g2 * D1 * ir3;
    gx += cu * ux + ci * mix + cj * mjx;
    gy += cu * uy + ci * miy + cj * mjy;
    gz += cu * uz + ci * miz + cj * mjz;
  }
  long long base = OFF_F + (((long long)(b * N + i)) * C + c) * 3;
  out[base + 0] = -CHG_FACTOR_F * gx;
  out[base + 1] = -CHG_FACTOR_F * gy;
  out[base + 2] = -CHG_FACTOR_F * gz;
}

extern "C" void kernel_launch(void* const* d_in, const int* in_sizes, int n_in,
                              void* d_out, int out_size, void* d_ws, size_t ws_size,
                              hipStream_t stream) {
  const float* coords = (const float*)d_in[0];
  const float* alpha  = (const float*)d_in[1];
  const float* charge = (const float*)d_in[2];
  const int* n12 = (const int*)d_in[4];
  const int* n13 = (const int*)d_in[5];
  const int* n14 = (const int*)d_in[6];
  const int* n15 = (const int*)d_in[7];
  const float* ind14 = (const float*)d_in[8];
  const float* ind15 = (const float*)d_in[9];
  float* out = (float*)d_out;
  float* ws  = (float*)d_ws;
  float* sc  = ws + WS_SC;
  float* Aw  = ws + WS_A;
  float* iLw = ws + WS_IL;
  float* muw = ws + WS_MU;

  pol_scaler_init<<<(int)((B * NN + 255) / 256), 256, 0, stream>>>(sc);
  int np;
  np = in_sizes[4] / 2;
  pol_scaler_pairs<<<(np + 255) / 256, 256, 0, stream>>>(sc, n12, np, nullptr);
  np = in_sizes[5] / 2;
  pol_scaler_pairs<<<(np + 255) / 256, 256, 0, stream>>>(sc, n13, np, nullptr);
  np = in_sizes[6] / 2;
  pol_scaler_pairs<<<(np + 255) / 256, 256, 0, stream>>>(sc, n14, np, ind14);
  np = in_sizes[7] / 2;
  pol_scaler_pairs<<<(np + 255) / 256, 256, 0, stream>>>(sc, n15, np, ind15);

  pol_build<<<(int)((B * C * NN) / 256), 256, 0, stream>>>(coords, alpha, sc, out, Aw);
  pol_ep<<<(B * C * N + 255) / 256, 256, 0, stream>>>(coords, alpha, charge, sc, out);
  pol_chol<<<NS, 256, 0, stream>>>(Aw, iLw);
  pol_solve<<<NS, 256, 0, stream>>>(Aw, iLw, out, muw);
  pol_grad<<<(B * C * N + 255) / 256, 256, 0, stream>>>(coords, alpha, charge, sc, muw, out);
}